// GATModel_26723286516177
// MI455X (gfx1250) — compile-verified
//
#include <hip/hip_runtime.h>
#include <hip/hip_bf16.h>
#include <string.h>
#include <stdint.h>

// ---------------- problem constants ----------------
#define Bg      128          // graphs
#define NPG     512          // nodes per graph
#define NTOT    (Bg * NPG)   // 65536 nodes
#define EG      504          // edges per graph
#define EREAL   (Bg * EG)    // 64512 real edges
#define EMAX    (EREAL + NTOT)
#define HH      8            // heads
#define CC      64           // channels per head
#define FF      (HH * CC)    // 512
#define AGENTS  8
#define OUTD    2
#define NEGS    0.2f

typedef float v2f __attribute__((ext_vector_type(2)));
typedef float v8f __attribute__((ext_vector_type(8)));

// ---------------- helpers ----------------
__device__ __forceinline__ unsigned fkey(float f) {
    unsigned u = __float_as_uint(f);
    return (u & 0x80000000u) ? ~u : (u | 0x80000000u);
}
__device__ __forceinline__ float funkey(unsigned k) {
    return __uint_as_float((k & 0x80000000u) ? (k ^ 0x80000000u) : ~k);
}
__device__ __forceinline__ void edge_sd(const int* __restrict__ ei, int e, int Ereal,
                                        int& s, int& d) {
    if (e < Ereal) {
        int boff = (e / EG) * NPG;
        s = ei[2 * e]     + boff;
        d = ei[2 * e + 1] + boff;
    } else {
        s = d = e - Ereal;
    }
}

// CDNA5 async copy: global -> LDS, 16 bytes per lane, tracked by ASYNCcnt.
// VGLOBAL async encoding (ISA 15.18.3 op 98): vdst = LDS byte address VGPR,
// vaddr = 64-bit global address. LDS offset = low 32 bits of flat shared ptr
// (aperture rule).
__device__ __forceinline__ void async_ld16(unsigned lds_off, const float* g) {
    asm volatile("global_load_async_to_lds_b128 %0, %1, off"
                 :: "v"(lds_off), "v"(g) : "memory");
}
__device__ __forceinline__ unsigned lds_addr32(const void* p) {
    return (unsigned)(uintptr_t)p;
}

// ---------------- generic fills ----------------
__global__ void fill_f32(float* __restrict__ p, float v, int n) {
    int i = blockIdx.x * blockDim.x + threadIdx.x;
    if (i < n) p[i] = v;
}
__global__ void fill_u32(unsigned* __restrict__ p, unsigned v, int n) {
    int i = blockIdx.x * blockDim.x + threadIdx.x;
    if (i < n) p[i] = v;
}

// ---------------- layer-1 linear (K=3) ----------------
__global__ void gemm_k3(const float* __restrict__ x, const float* __restrict__ W,
                        float* __restrict__ xs) {
    int idx = blockIdx.x * blockDim.x + threadIdx.x;  // over NTOT*FF
    if (idx >= NTOT * FF) return;
    int n = idx >> 9, f = idx & 511;
    const float* xr = x + (long)n * 3;
    xs[idx] = xr[0] * W[f] + xr[1] * W[FF + f] + xr[2] * W[2 * FF + f];
}

// ---------------- WMMA f32 GEMM: C[M,N] = A[M,K] @ B[K,N] ----------------
// 256 threads = 8 waves stacked in M. Block tile 128x64, wave tile 16x64
// (4 accumulators -> A-frag reused 4x). K-step 16. A/B tiles staged in LDS
// via global_load_async_to_lds_b128 with ASYNCcnt double buffering.
__global__ __launch_bounds__(256) void wmma_gemm_f32(
    const float* __restrict__ A, const float* __restrict__ Bm, float* __restrict__ Cm,
    int M, int Nn, int K) {
    __shared__ float lA[2][128 * 16];
    __shared__ float lB[2][16 * 64];
    const int t = threadIdx.x;
    const int wave = t >> 5, lane = t & 31;
    const int mBase = blockIdx.y * 128;
    const int nBase = blockIdx.x * 64;
    const int half = (lane >> 4) & 1;   // lanes 16-31: K+2 (A/B frags), M+8 (C/D)
    const int l15  = lane & 15;

    // per-thread async staging coordinates
    const int ra  = t >> 2, ca = (t & 3) * 4;    // A tile: 128x16, 2 x b128/thread
    const int rb  = t >> 4, cb = (t & 15) * 4;   // B tile: 16x64,  1 x b128/thread

    v8f acc[4];
#pragma unroll
    for (int ns = 0; ns < 4; ++ns)
        acc[ns] = (v8f){0.f, 0.f, 0.f, 0.f, 0.f, 0.f, 0.f, 0.f};

    const int nK = K >> 4;

    auto issue = [&](int kt, int bi) {
        const int kk = kt << 4;
        unsigned aB = lds_addr32(&lA[bi][0]);
        unsigned bB = lds_addr32(&lB[bi][0]);
        async_ld16(aB + (unsigned)((ra * 16 + ca) * 4),
                   A + (long)(mBase + ra) * K + kk + ca);
        async_ld16(aB + (unsigned)(((ra + 64) * 16 + ca) * 4),
                   A + (long)(mBase + ra + 64) * K + kk + ca);
        async_ld16(bB + (unsigned)((rb * 64 + cb) * 4),
                   Bm + (long)(kk + rb) * Nn + nBase + cb);
    };

    issue(0, 0);
    int cur = 0;
    for (int kt = 0; kt < nK; ++kt) {
        if (kt + 1 < nK) {
            issue(kt + 1, cur ^ 1);                       // other buffer: released
            asm volatile("s_wait_asynccnt 0x3" ::: "memory");  // tile kt landed
        } else {
            asm volatile("s_wait_asynccnt 0x0" ::: "memory");
        }
        __syncthreads();
        const float* __restrict__ tA = &lA[cur][0];
        const float* __restrict__ tB = &lB[cur][0];
#pragma unroll
        for (int k4 = 0; k4 < 16; k4 += 4) {
            const int kb = k4 + 2 * half;
            v2f af;
            af.x = tA[(wave * 16 + l15) * 16 + kb];
            af.y = tA[(wave * 16 + l15) * 16 + kb + 1];
#pragma unroll
            for (int ns = 0; ns < 4; ++ns) {
                v2f bf;
                bf.x = tB[kb * 64 + ns * 16 + l15];
                bf.y = tB[(kb + 1) * 64 + ns * 16 + l15];
                acc[ns] = __builtin_amdgcn_wmma_f32_16x16x4_f32(
                    false, af, false, bf, (short)0, acc[ns], false, false);
            }
        }
        __syncthreads();   // release buffer `cur` for tile kt+2
        cur ^= 1;
    }

#pragma unroll
    for (int ns = 0; ns < 4; ++ns) {
        const int col = nBase + ns * 16 + l15;
#pragma unroll
        for (int v = 0; v < 8; ++v) {
            int row = mBase + wave * 16 + v + 8 * half;
            Cm[(long)row * Nn + col] = acc[ns][v];
        }
    }
}

// ---------------- per-node attention scalars ----------------
__global__ void attn_node(const float* __restrict__ xs, const float* __restrict__ a_s,
                          const float* __restrict__ a_d, float* __restrict__ als,
                          float* __restrict__ ald) {
    int idx = blockIdx.x * blockDim.x + threadIdx.x;  // over NTOT*HH
    if (idx >= NTOT * HH) return;
    int n = idx >> 3, h = idx & 7;
    const float* xr = xs + (long)n * FF + h * CC;
    const float* as = a_s + h * CC;
    const float* ad = a_d + h * CC;
    float ss = 0.f, sd = 0.f;
#pragma unroll 8
    for (int c = 0; c < CC; ++c) {
        float v = xr[c];
        ss += v * as[c];
        sd += v * ad[c];
    }
    als[idx] = ss;
    ald[idx] = sd;
}

// ---------------- edge-attr constant: ce[h] = sum_c We[h*64+c]*ae[h*64+c] ---
__global__ void ce_kernel(const float* __restrict__ We, const float* __restrict__ ae,
                          float* __restrict__ ce) {
    int h = threadIdx.x;
    if (h >= HH) return;
    float s = 0.f;
    for (int c = 0; c < CC; ++c) s += We[h * CC + c] * ae[h * CC + c];
    ce[h] = s;
}

// ---------------- self-loop attr = mean of incoming edge attrs -------------
__global__ void loop_accum(const int* __restrict__ ei, const float* __restrict__ ea,
                           float* __restrict__ lsum, float* __restrict__ lcnt) {
    int e = blockIdx.x * blockDim.x + threadIdx.x;
    if (e >= EREAL) return;
    int s, d;
    edge_sd(ei, e, EREAL, s, d);
    atomicAdd(&lsum[d], ea[e]);
    atomicAdd(&lcnt[d], 1.0f);
}
__global__ void loop_final(const float* __restrict__ lsum, const float* __restrict__ lcnt,
                           float* __restrict__ lattr) {
    int n = blockIdx.x * blockDim.x + threadIdx.x;
    if (n >= NTOT) return;
    lattr[n] = lsum[n] / fmaxf(lcnt[n], 1.0f);
}

// ---------------- edge softmax passes ----------------
__global__ void alpha_raw(const int* __restrict__ ei, const float* __restrict__ ea,
                          const float* __restrict__ lattr, const float* __restrict__ als,
                          const float* __restrict__ ald, const float* __restrict__ ce,
                          float* __restrict__ alpha, unsigned* __restrict__ segmax,
                          int Etot) {
    int idx = blockIdx.x * blockDim.x + threadIdx.x;
    if (idx >= Etot * HH) return;
    int e = idx >> 3, h = idx & 7;
    int s, d;
    edge_sd(ei, e, EREAL, s, d);
    float av = (e < EREAL) ? ea[e] : lattr[e - EREAL];
    float v = als[s * HH + h] + ald[d * HH + h] + av * ce[h];
    v = (v > 0.f) ? v : NEGS * v;
    alpha[idx] = v;
    atomicMax(&segmax[d * HH + h], fkey(v));
}
__global__ void alpha_exp(const int* __restrict__ ei, float* __restrict__ alpha,
                          const unsigned* __restrict__ segmax, float* __restrict__ segsum,
                          int Etot) {
    int idx = blockIdx.x * blockDim.x + threadIdx.x;
    if (idx >= Etot * HH) return;
    int e = idx >> 3, h = idx & 7;
    int s, d;
    edge_sd(ei, e, EREAL, s, d);
    float v = expf(alpha[idx] - funkey(segmax[d * HH + h]));
    alpha[idx] = v;
    atomicAdd(&segsum[d * HH + h], v);
}
__global__ void alpha_norm(const int* __restrict__ ei, float* __restrict__ alpha,
                           const float* __restrict__ segsum, int Etot) {
    int idx = blockIdx.x * blockDim.x + threadIdx.x;
    if (idx >= Etot * HH) return;
    int e = idx >> 3, h = idx & 7;
    int s, d;
    edge_sd(ei, e, EREAL, s, d);
    alpha[idx] /= fmaxf(segsum[d * HH + h], 1e-16f);
}

// ---------------- aggregation: one wave32 per edge ----------------
__global__ __launch_bounds__(256) void aggregate(const int* __restrict__ ei,
                                                 const float* __restrict__ xs,
                                                 const float* __restrict__ alpha,
                                                 float* __restrict__ out, int Etot) {
    int wave = threadIdx.x >> 5;
    int lane = threadIdx.x & 31;
    int e = blockIdx.x * 8 + wave;
    if (e >= Etot) return;
    int s, d;
    edge_sd(ei, e, EREAL, s, d);
    const float* xr = xs + (long)s * FF;
    float* orow = out + (long)d * FF;
#pragma unroll
    for (int i = 0; i < 16; ++i) {
        int c = i * 32 + lane;
        float w = alpha[e * HH + (i >> 1)];  // head uniform across wave per iter
        atomicAdd(&orow[c], w * xr[c]);
    }
}

// ---------------- bias + relu (in place) ----------------
__global__ void bias_relu(float* __restrict__ p, const float* __restrict__ bias,
                          int total, int fdim) {
    int idx = blockIdx.x * blockDim.x + threadIdx.x;
    if (idx >= total) return;
    float v = p[idx] + bias[idx % fdim];
    p[idx] = fmaxf(v, 0.f);
}

// ---------------- graph mean pool ----------------
__global__ void pool_mean(const float* __restrict__ h, float* __restrict__ gemb) {
    int idx = blockIdx.x * blockDim.x + threadIdx.x;  // over Bg*FF
    if (idx >= Bg * FF) return;
    int b = idx >> 9, f = idx & 511;
    const float* base = h + ((long)b * NPG) * FF + f;
    float s = 0.f;
    for (int n = 0; n < NPG; ++n) s += base[(long)n * FF];
    gemb[idx] = s * (1.0f / NPG);
}

// ---------------- build [agent_emb | graph_emb] ----------------
__global__ void build_comb(const float* __restrict__ h, const float* __restrict__ gemb,
                           float* __restrict__ comb) {
    int idx = blockIdx.x * blockDim.x + threadIdx.x;  // over 1024*1024
    if (idx >= Bg * AGENTS * 2 * FF) return;
    int r = idx >> 10, f2 = idx & 1023;
    int b = r >> 3, a = r & 7;
    comb[idx] = (f2 < FF) ? h[((long)b * NPG + a) * FF + f2]
                          : gemb[b * FF + (f2 - FF)];
}

// ---------------- fc2 ----------------
__global__ void fc2_kernel(const float* __restrict__ hid, const float* __restrict__ w,
                           const float* __restrict__ bias, float* __restrict__ out) {
    int idx = blockIdx.x * blockDim.x + threadIdx.x;  // over 1024*OUTD
    if (idx >= Bg * AGENTS * OUTD) return;
    int r = idx >> 1, o = idx & 1;
    const float* hr = hid + (long)r * 256;
    float s = bias[o];
    for (int k = 0; k < 256; ++k) s += hr[k] * w[k * OUTD + o];
    out[idx] = s;
}

// ---------------- launcher ----------------
extern "C" void kernel_launch(void* const* d_in, const int* in_sizes, int n_in,
                              void* d_out, int out_size, void* d_ws, size_t ws_size,
                              hipStream_t stream) {
    const float* x  = (const float*)d_in[0];
    const int*   ei = (const int*)d_in[1];
    const float* ea = (const float*)d_in[2];
    const float* W[3]  = {(const float*)d_in[3],  (const float*)d_in[9],  (const float*)d_in[15]};
    const float* bb[3] = {(const float*)d_in[4],  (const float*)d_in[10], (const float*)d_in[16]};
    const float* as_[3]= {(const float*)d_in[5],  (const float*)d_in[11], (const float*)d_in[17]};
    const float* ad_[3]= {(const float*)d_in[6],  (const float*)d_in[12], (const float*)d_in[18]};
    const float* We[3] = {(const float*)d_in[7],  (const float*)d_in[13], (const float*)d_in[19]};
    const float* ae[3] = {(const float*)d_in[8],  (const float*)d_in[14], (const float*)d_in[20]};
    const float* fc1w = (const float*)d_in[21];
    const float* fc1b = (const float*)d_in[22];
    const float* fc2w = (const float*)d_in[23];
    const float* fc2b = (const float*)d_in[24];
    float* outp = (float*)d_out;

    // workspace carve-up (all chunk sizes are multiples of 4 floats -> 16B aligned)
    float* ws = (float*)d_ws;
    size_t off = 0;
    auto alloc = [&](size_t n) { float* p = ws + off; off += n; return p; };
    const size_t NF = (size_t)NTOT * FF;
    float*    bufA   = alloc(NF);         // layer input / aggregation output
    float*    bufB   = alloc(NF);         // xs = linear(input)
    float*    als    = alloc((size_t)NTOT * HH);
    float*    ald    = alloc((size_t)NTOT * HH);
    float*    alpha  = alloc((size_t)EMAX * HH);
    unsigned* segmax = (unsigned*)alloc((size_t)NTOT * HH);
    float*    segsum = alloc((size_t)NTOT * HH);
    float*    lattr  = alloc(NTOT);
    float*    lsum   = alloc(NTOT);
    float*    lcnt   = alloc(NTOT);
    float*    ceArr  = alloc(16);
    float*    gemb   = alloc((size_t)Bg * FF);
    float*    comb   = alloc((size_t)Bg * AGENTS * 2 * FF);
    float*    hid    = alloc((size_t)Bg * AGENTS * 4 * 64);

    // host-side monotonic key for -1e30 (segmax init)
    float ninf = -1e30f;
    unsigned nbits;
    memcpy(&nbits, &ninf, 4);
    unsigned ninfKey = (nbits & 0x80000000u) ? ~nbits : (nbits | 0x80000000u);

    const int TPB = 256;
    auto blks = [](long n, int t) { return (unsigned)((n + t - 1) / t); };

    for (int l = 0; l < 3; ++l) {
        const bool selfLoops = (l > 0);
        const int Etot = selfLoops ? EMAX : EREAL;

        // 1) linear transform -> bufB
        if (l == 0) {
            gemm_k3<<<blks((long)NTOT * FF, TPB), TPB, 0, stream>>>(x, W[0], bufB);
        } else {
            wmma_gemm_f32<<<dim3(FF / 64, NTOT / 128), 256, 0, stream>>>(
                bufA, W[l], bufB, NTOT, FF, FF);
        }

        // 2) attention scalars
        attn_node<<<blks((long)NTOT * HH, TPB), TPB, 0, stream>>>(bufB, as_[l], ad_[l], als, ald);
        ce_kernel<<<1, 32, 0, stream>>>(We[l], ae[l], ceArr);

        // 3) self-loop attrs
        if (selfLoops) {
            fill_f32<<<blks(NTOT, TPB), TPB, 0, stream>>>(lsum, 0.f, NTOT);
            fill_f32<<<blks(NTOT, TPB), TPB, 0, stream>>>(lcnt, 0.f, NTOT);
            loop_accum<<<blks(EREAL, TPB), TPB, 0, stream>>>(ei, ea, lsum, lcnt);
            loop_final<<<blks(NTOT, TPB), TPB, 0, stream>>>(lsum, lcnt, lattr);
        }

        // 4) edge softmax
        fill_u32<<<blks((long)NTOT * HH, TPB), TPB, 0, stream>>>(segmax, ninfKey, NTOT * HH);
        fill_f32<<<blks((long)NTOT * HH, TPB), TPB, 0, stream>>>(segsum, 0.f, NTOT * HH);
        alpha_raw<<<blks((long)Etot * HH, TPB), TPB, 0, stream>>>(
            ei, ea, lattr, als, ald, ceArr, alpha, segmax, Etot);
        alpha_exp<<<blks((long)Etot * HH, TPB), TPB, 0, stream>>>(ei, alpha, segmax, segsum, Etot);
        alpha_norm<<<blks((long)Etot * HH, TPB), TPB, 0, stream>>>(ei, alpha, segsum, Etot);

        // 5) aggregate into bufA (zero first)
        fill_f32<<<blks((long)NF, TPB), TPB, 0, stream>>>(bufA, 0.f, (int)NF);
        aggregate<<<blks(Etot, 8), 256, 0, stream>>>(ei, bufB, alpha, bufA, Etot);

        // 6) bias + relu
        bias_relu<<<blks((long)NF, TPB), TPB, 0, stream>>>(bufA, bb[l], (int)NF, FF);
    }

    // head
    pool_mean<<<blks((long)Bg * FF, TPB), TPB, 0, stream>>>(bufA, gemb);
    build_comb<<<blks((long)Bg * AGENTS * 2 * FF, TPB), TPB, 0, stream>>>(bufA, gemb, comb);
    wmma_gemm_f32<<<dim3(256 / 64, (Bg * AGENTS) / 128), 256, 0, stream>>>(
        comb, fc1w, hid, Bg * AGENTS, 256, 2 * FF);
    bias_relu<<<blks((long)Bg * AGENTS * 256, TPB), TPB, 0, stream>>>(
        hid, fc1b, Bg * AGENTS * 256, 256);
    fc2_kernel<<<blks((long)Bg * AGENTS * OUTD, TPB), TPB, 0, stream>>>(hid, fc2w, fc2b, outp);
}